// dsfa_loss_43267500540053
// MI455X (gfx1250) — compile-verified
//
#include <hip/hip_runtime.h>

// sum(sigma^2) over 1048576x128 f32 -> scalar f32.
// Bandwidth-bound streaming reduction: 512 MiB read @ 23.3 TB/s ~= 22.5 us
// floor; FLOPs (268M) are irrelevant. Strategy:
//  - non-temporal global_load_b128 x4 per iteration (64 B/thread/iter,
//    one base-address add, immediate offsets 0/16/32/48)
//  - uniform trip count -> scalar loop control, no per-iteration EXEC churn
//  - 8 independent accumulators -> no FMA dependency stalls
//  - wave reduction via V_WMMA_F32_16X16X4_F32 with B = ones
//  - deterministic two-pass finish (no float atomics)

typedef float v2f __attribute__((ext_vector_type(2)));
typedef float v4f __attribute__((ext_vector_type(4)));
typedef float v8f __attribute__((ext_vector_type(8)));

// Reduce two f32 partials per lane across the whole wave32 using one WMMA.
// A (16x4 f32) = 2 VGPRs x 32 lanes = the wave's 64 partials.
// B (4x16 f32) = all ones  =>  D[m][n] = sum_k A[m][k] (row sums, replicated
// across columns). Summing D's 8 VGPRs in-lane gives rows 0-7 total in lanes
// 0-15 and rows 8-15 total in lanes 16-31; one half-swap + add completes the
// wave sum. All f32 precision.
__device__ __forceinline__ float wave_reduce_wmma(float x, float y) {
    v2f a;
    a.x = x;
    a.y = y;
    v2f b;
    b.x = 1.0f;
    b.y = 1.0f;
    v8f c = {};
    c = __builtin_amdgcn_wmma_f32_16x16x4_f32(
        /*neg_a=*/false, a, /*neg_b=*/false, b,
        /*c_mod=*/(short)0, c, /*reuse_a=*/false, /*reuse_b=*/false);
    float h = c[0] + c[1] + c[2] + c[3] + c[4] + c[5] + c[6] + c[7];
    h += __shfl_xor(h, 16, 32);   // lower-half total + upper-half total
    return h;                      // full wave sum in every lane
}

// Block-level reduction: WMMA per wave, LDS combine. Result valid in thread 0.
// Must be reached uniformly by all threads (EXEC all-ones for WMMA).
__device__ __forceinline__ float block_reduce(float x, float y) {
    __shared__ float smem[32];
    const float w = wave_reduce_wmma(x, y);
    const int lane = threadIdx.x & 31;
    const int wid  = threadIdx.x >> 5;
    if (lane == 0) smem[wid] = w;
    __syncthreads();
    float total = 0.0f;
    if (threadIdx.x == 0) {
        const int nwaves = (blockDim.x + 31) >> 5;
        for (int i = 0; i < nwaves; ++i) total += smem[i];
    }
    return total;
}

__device__ __forceinline__ void acc_sq(const v4f v, float& e, float& o) {
    e = fmaf(v.x, v.x, e);
    o = fmaf(v.y, v.y, o);
    e = fmaf(v.z, v.z, e);
    o = fmaf(v.w, v.w, o);
}

// Pass 1: each thread consumes `iters` chunks of 4 consecutive float4s (64 B),
// chunks strided by the grid. One f32 partial per block to workspace.
__global__ void sumsq_partial_kernel(const v4f* __restrict__ in,
                                     float* __restrict__ partials,
                                     long long n4) {
    const long long nthreads = (long long)gridDim.x * blockDim.x;
    const long long tid      = (long long)blockIdx.x * blockDim.x + threadIdx.x;

    // Uniform (thread-invariant) trip count: scalar loop control.
    const long long chunk_stride = nthreads * 4;        // float4s per iteration
    const long long iters        = n4 / chunk_stride;   // same for every thread

    float a0 = 0.f, a1 = 0.f, a2 = 0.f, a3 = 0.f;
    float a4 = 0.f, a5 = 0.f, a6 = 0.f, a7 = 0.f;

    const v4f* p = in + tid * 4;
    for (long long k = 0; k < iters; ++k) {
        const v4f v0 = __builtin_nontemporal_load(p + 0);  // b128 nt, off 0
        const v4f v1 = __builtin_nontemporal_load(p + 1);  // off 16
        const v4f v2 = __builtin_nontemporal_load(p + 2);  // off 32
        const v4f v3 = __builtin_nontemporal_load(p + 3);  // off 48
        p += chunk_stride;
        acc_sq(v0, a0, a1);
        acc_sq(v1, a2, a3);
        acc_sq(v2, a4, a5);
        acc_sq(v3, a6, a7);
    }

    // Tail: remaining n4 % chunk_stride float4s, one per thread max per step.
    for (long long i = iters * chunk_stride + tid; i < n4; i += nthreads) {
        const v4f v = __builtin_nontemporal_load(in + i);
        acc_sq(v, a0, a1);
    }
    // (reconverged here: EXEC all-ones for the WMMA below)

    const float x = (a0 + a2) + (a4 + a6);
    const float y = (a1 + a3) + (a5 + a7);
    const float t = block_reduce(x, y);
    if (threadIdx.x == 0) partials[blockIdx.x] = t;
}

// Pass 2: single block folds per-block partials into d_out[0].
__global__ void sumsq_final_kernel(const float* __restrict__ partials,
                                   float* __restrict__ out,
                                   int nparts) {
    float acc = 0.0f;
    for (int i = threadIdx.x; i < nparts; i += blockDim.x) acc += partials[i];
    const float t = block_reduce(acc, 0.0f);
    if (threadIdx.x == 0) out[0] = t;
}

extern "C" void kernel_launch(void* const* d_in, const int* in_sizes, int n_in,
                              void* d_out, int out_size, void* d_ws, size_t ws_size,
                              hipStream_t stream) {
    (void)n_in; (void)out_size; (void)ws_size;

    const v4f* in = (const v4f*)d_in[0];
    float* ws     = (float*)d_ws;
    float* out    = (float*)d_out;

    const long long n  = (long long)in_sizes[0];   // 1048576 * 128 (mult. of 4)
    const long long n4 = n >> 2;

    const int threads = 256;   // 8 wave32s per block
    const int blocks  = 2048;  // 8 KB of f32 partials in d_ws

    sumsq_partial_kernel<<<blocks, threads, 0, stream>>>(in, ws, n4);
    sumsq_final_kernel<<<1, threads, 0, stream>>>(ws, out, blocks);
}